// loss_kernelMatching_mask_batch_14499809591887
// MI455X (gfx1250) — compile-verified
//
#include <hip/hip_runtime.h>
#include <hip/hip_bf16.h>

// Masked cosine loss, bs=16, N=4096 (64x64), D=512, fp32.
// Memory-bound: 268 MB streamed once -> ~11.5 us at 23.3 TB/s.
// Per-wave 16-row tiles; per-row sums = diagonals of P.T^T / P.P^T / T.T^T
// via V_WMMA_F32_16X16X4_F32 (full fp32). The f32 16x4 A-layout and 4x16
// B-layout coincide per (lane,vgpr) and only the diagonal is consumed, so
// contiguous b128 loads feed WMMA with zero cross-lane shuffles.
// v2: 2-deep software pipeline (~4 loads in flight/wave) + KSPLIT=4
// (16384 waves) for deeper memory-level parallelism. Deterministic
// two-stage reduction (no atomics).

typedef __attribute__((ext_vector_type(2))) float v2f;
typedef __attribute__((ext_vector_type(4))) float v4f;
typedef __attribute__((ext_vector_type(8))) float v8f;

#define BSZ     16
#define NROWS   4096
#define DDIM    512
#define KSPLIT  4                    // waves per 16-row group (split over D)
#define WPB     (256 * KSPLIT)       // waves per batch = 1024
#define TOTW    (BSZ * WPB)          // total waves     = 16384
// workspace: TOTW * 3 floats = 192 KB

__device__ __forceinline__ float wave_sum(float v) {
  #pragma unroll
  for (int off = 16; off > 0; off >>= 1) v += __shfl_xor(v, off, 32);
  return v;
}

#define WMMA6(pa, ta)                                                                              \
  do {                                                                                             \
    v2f p0 = { (pa)[0], (pa)[1] }, p1 = { (pa)[2], (pa)[3] };                                      \
    v2f t0 = { (ta)[0], (ta)[1] }, t1 = { (ta)[2], (ta)[3] };                                      \
    accPT = __builtin_amdgcn_wmma_f32_16x16x4_f32(false, p0, false, t0, (short)0, accPT, false, false); \
    accPP = __builtin_amdgcn_wmma_f32_16x16x4_f32(false, p0, false, p0, (short)0, accPP, false, false); \
    accTT = __builtin_amdgcn_wmma_f32_16x16x4_f32(false, t0, false, t0, (short)0, accTT, false, false); \
    accPT = __builtin_amdgcn_wmma_f32_16x16x4_f32(false, p1, false, t1, (short)0, accPT, false, false); \
    accPP = __builtin_amdgcn_wmma_f32_16x16x4_f32(false, p1, false, p1, (short)0, accPP, false, false); \
    accTT = __builtin_amdgcn_wmma_f32_16x16x4_f32(false, t1, false, t1, (short)0, accTT, false, false); \
  } while (0)

__global__ __launch_bounds__(256)
void mcos_partials(const float* __restrict__ pred,
                   const float* __restrict__ tgt,
                   const int*   __restrict__ mask,
                   float*       __restrict__ ws) {
  const int tid  = blockIdx.x * blockDim.x + threadIdx.x;
  const int gw   = tid >> 5;            // global wave id
  const int lane = threadIdx.x & 31;

  const int b    = gw / WPB;            // batch
  const int wb   = gw % WPB;
  const int rgrp = wb / KSPLIT;         // 16-row group within batch
  const int ks   = wb % KSPLIT;         // D-split index
  const int nbase = rgrp * 16;

  const int r16 = lane & 15;            // row within group (A/B layouts agree)
  const int h   = lane >> 4;            // lane half selects k sub-slot

  const size_t rowoff = ((size_t)b * NROWS + nbase + r16) * DDIM;
  const int    dstart = ks * (DDIM / KSPLIT);
  const float* P = pred + rowoff + dstart + 4 * h;
  const float* T = tgt  + rowoff + dstart + 4 * h;

  v8f accPT = {}; v8f accPP = {}; v8f accTT = {};

  const int iters = (DDIM / KSPLIT) / 8;   // 16: 8 consecutive floats/row/iter

  // 2-deep software pipeline: keep ~4 b128 loads in flight per wave.
  v4f pa = *(const v4f*)(P);
  v4f ta = *(const v4f*)(T);
  v4f pb = *(const v4f*)(P + 8);
  v4f tb = *(const v4f*)(T + 8);

  #pragma unroll 2
  for (int it = 0; it < iters - 2; ++it) {
    v4f pc = *(const v4f*)(P + (it + 2) * 8);
    v4f tc = *(const v4f*)(T + (it + 2) * 8);
    WMMA6(pa, ta);
    pa = pb; ta = tb;
    pb = pc; tb = tc;
  }
  WMMA6(pa, ta);
  WMMA6(pb, tb);

  // Diagonal of 16x16 f32 C/D: element (i,i) lives at
  //   i < 8  : vgpr i,     lane i
  //   i >= 8 : vgpr i - 8, lane i + 16
  const bool valid = (lane < 8) || (lane >= 24);
  const int  sel   = (lane < 8) ? lane : (lane - 24);  // negative -> no match
  const int  row   = (lane < 8) ? lane : (lane - 16);

  float dpt = 0.f, dpp = 0.f, dtt = 0.f;
  #pragma unroll
  for (int i = 0; i < 8; ++i) {
    if (sel == i) { dpt = accPT[i]; dpp = accPP[i]; dtt = accTT[i]; }
  }

  float m = 0.f;
  if (valid) m = (float)mask[(size_t)b * NROWS + nbase + row];
  dpt *= m; dpp *= m; dtt *= m;

  dpt = wave_sum(dpt);
  dpp = wave_sum(dpp);
  dtt = wave_sum(dtt);

  if (lane == 0) {
    float* o = ws + (size_t)gw * 3;
    o[0] = dpt; o[1] = dpp; o[2] = dtt;
  }
}

__global__ __launch_bounds__(256)
void mcos_finalize(const float* __restrict__ ws, float* __restrict__ out) {
  __shared__ float red[8][3];
  const int t = threadIdx.x, lane = t & 31, w = t >> 5;
  float total = 0.f;

  for (int b = 0; b < BSZ; ++b) {
    float pt = 0.f, pp = 0.f, tt = 0.f;
    for (int c = t; c < WPB; c += 256) {
      const float* o = ws + (size_t)(b * WPB + c) * 3;
      pt += o[0]; pp += o[1]; tt += o[2];
    }
    pt = wave_sum(pt); pp = wave_sum(pp); tt = wave_sum(tt);
    if (lane == 0) { red[w][0] = pt; red[w][1] = pp; red[w][2] = tt; }
    __syncthreads();
    if (t == 0) {
      float spt = 0.f, spp = 0.f, stt = 0.f;
      #pragma unroll
      for (int i = 0; i < 8; ++i) { spt += red[i][0]; spp += red[i][1]; stt += red[i][2]; }
      const float denom = sqrtf(spp) * sqrtf(stt);
      total += (denom > 0.f) ? (-spt / denom) : 0.f;
    }
    __syncthreads();
  }
  if (t == 0) out[0] = total / (float)BSZ;
}

extern "C" void kernel_launch(void* const* d_in, const int* in_sizes, int n_in,
                              void* d_out, int out_size, void* d_ws, size_t ws_size,
                              hipStream_t stream) {
  const float* pred = (const float*)d_in[0];   // [16, 4096, 512] f32
  const float* tgt  = (const float*)d_in[1];   // [16, 4096, 512] f32
  const int*   mask = (const int*)d_in[2];     // [16, 64, 64]    i32
  float* out = (float*)d_out;                  // scalar f32
  float* ws  = (float*)d_ws;                   // needs TOTW*3*4 = 192 KB

  const int blocks = (TOTW * 32) / 256;        // 2048 blocks x 8 waves
  mcos_partials<<<blocks, 256, 0, stream>>>(pred, tgt, mask, ws);
  mcos_finalize<<<1, 256, 0, stream>>>(ws, out);
}